// BandSplitDCTFilter_69329362092336
// MI455X (gfx1250) — compile-verified
//
#include <hip/hip_runtime.h>

// ---------------------------------------------------------------------------
// BandSplitDCTFilter for MI455X (gfx1250, wave32, WMMA f32 16x16x4, TDM)
//
// Math (derived from the reference's FFT-based DCT):
//   D    = Tc · X · Tc^T                      (pure cosine DCT-II sums)
//   E    = (g_kw/4096) · M' ⊙ D               g_0=1, g_{kw>0}=2
//   M'   = W_high + [kh<32,kw<32]·W_mid + [kh<16,kw<16]·W_low
//   xo   = Tc^T·E·Tc − Ts^T·E·Ts              (reference idct keeps only the
//                                              kh half-spectrum -> cos(a+b))
//   y    = xo · proj_w^T ;  out = LayerNorm_c(y)·g + b
// ---------------------------------------------------------------------------

#if defined(__HIP_DEVICE_COMPILE__) && \
    !__has_builtin(__builtin_amdgcn_wmma_f32_16x16x4_f32)
#error "gfx1250 f32 WMMA builtin __builtin_amdgcn_wmma_f32_16x16x4_f32 not available"
#endif

typedef float v2f __attribute__((ext_vector_type(2)));
typedef float v8f __attribute__((ext_vector_type(8)));
typedef unsigned int u32x4 __attribute__((ext_vector_type(4)));
typedef int i32x4 __attribute__((ext_vector_type(4)));
typedef int i32x8 __attribute__((ext_vector_type(8)));

#define PI_F 3.14159265358979323846f

// Padded LDS row strides: stride % 64 == 4 banks-of-4B pattern ->
// for b64 loads lanes 0..15 hit banks {4n,4n+1}, lanes 16..31 (k+2) hit
// {4n+2,4n+3}: all 64 banks distinct, conflict-free.
#define S64  68    // padded stride for 64-wide rows
#define S128 132   // padded stride for 128-wide rows
#define S256 260   // padded stride for 256-wide rows

// Generic WMMA pass: D[Mtiles*16 x Ntiles*16] = A[.. x K] * B[K x ..]
// Each wave owns MGROUP m-tiles sharing a single n-tile: one B fragment feeds
// MGROUP WMMAs.  K is processed in chunks of KCHUNK: all chunk loads issue as
// a burst (deep dscnt pipelining), then KCHUNK*MGROUP back-to-back WMMAs.
// fa2(m,k0) -> v2f {A[m][k0],A[m][k0+1]}, fb2(k0,n) -> v2f {B[k0][n],B[k0+1][n]}
// fo(m,n,v) -> store D element.
// A-frag (16x4 f32): lane l holds A[m = l&15][k0 = 4ks + 2*(l>>4) + {0,1}]
// B-frag (4x16 f32): lane l holds B[same k0 pair][n = l&15]
// C/D  (16x16 f32): vgpr r, lane l -> M = r + 8*(l>>4), N = l&15
template <int MGROUP, int KCHUNK, typename FA2, typename FB2, typename FO>
__device__ __forceinline__ void wmma_pass(int Mtiles, int Ntiles, int Ksteps,
                                          FA2 fa2, FB2 fb2, FO fo) {
#if defined(__HIP_DEVICE_COMPILE__)
  const int lane = threadIdx.x & 31;
  const int wave = threadIdx.x >> 5;   // 16 waves per 512-thread block
  const int half = lane >> 4;
  const int lm   = lane & 15;
  const int mgroups = Mtiles / MGROUP;
  const int groups  = mgroups * Ntiles;
  for (int gi = wave; gi < groups; gi += 16) {
    const int mg = (gi % mgroups) * MGROUP;
    const int nb = (gi / mgroups) * 16;
    v8f acc[MGROUP];
#pragma unroll
    for (int mi = 0; mi < MGROUP; ++mi)
      acc[mi] = (v8f){0.f, 0.f, 0.f, 0.f, 0.f, 0.f, 0.f, 0.f};
    for (int kc = 0; kc < Ksteps; kc += KCHUNK) {
      v2f breg[KCHUNK];
      v2f areg[MGROUP][KCHUNK];
#pragma unroll
      for (int kk = 0; kk < KCHUNK; ++kk) {
        const int k0 = ((kc + kk) << 2) + (half << 1);
        breg[kk] = fb2(k0, nb + lm);
#pragma unroll
        for (int mi = 0; mi < MGROUP; ++mi)
          areg[mi][kk] = fa2((mg + mi) * 16 + lm, k0);
      }
#pragma unroll
      for (int kk = 0; kk < KCHUNK; ++kk)
#pragma unroll
        for (int mi = 0; mi < MGROUP; ++mi)
          acc[mi] = __builtin_amdgcn_wmma_f32_16x16x4_f32(
              false, areg[mi][kk], false, breg[kk],
              (short)0, acc[mi], false, false);
    }
#pragma unroll
    for (int mi = 0; mi < MGROUP; ++mi) {
      const int mb = (mg + mi) * 16;
#pragma unroll
      for (int r = 0; r < 8; ++r) fo(mb + r + half * 8, nb + lm, acc[mi][r]);
    }
  }
#endif
}

// ---------------------------------------------------------------------------
// Prep: transform matrices (unpadded in workspace).
//  ws[0..8191]     = TP3[m][k]  : m<64 -> Tc[k][m], m>=64 -> Ts[k][m-64]
//  ws[8192..16383] = TP4[j][k2] : k2<64 -> Tc[k2][j], else -Ts[k2-64][j]
__global__ void prep_T_kernel(float* ws) {
  int gid = blockIdx.x * 256 + threadIdx.x;
  if (gid < 8192) {
    int m = gid >> 6, k = gid & 63;
    float ang = PI_F * (float)k * (float)(2 * (m & 63) + 1) * (1.0f / 128.0f);
    ws[gid] = (m < 64) ? __cosf(ang) : __sinf(ang);
  } else if (gid < 16384) {
    int q = gid - 8192;
    int j = q >> 7, k2 = q & 127;
    float ang = PI_F * (float)(k2 & 63) * (float)(2 * j + 1) * (1.0f / 128.0f);
    ws[gid] = (k2 < 64) ? __cosf(ang) : -__sinf(ang);
  }
}

// Prep: combined filter  msc[kh][kw][c] = g_kw/4096 * M'
__global__ void prep_filter_kernel(const float* __restrict__ Wl,
                                   const float* __restrict__ Wm,
                                   const float* __restrict__ Wh,
                                   float* __restrict__ msc) {
  int gid = blockIdx.x * 256 + threadIdx.x;   // 64*64*256 total
  int c = gid & 255, kw = (gid >> 8) & 63, kh = gid >> 14;
  float v = Wh[gid];
  if (kh < 32 && kw < 32) v += Wm[(((kh << 5) + kw) << 8) + c];
  if (kh < 16 && kw < 16) v += Wl[(((kh << 4) + kw) << 8) + c];
  msc[gid] = v * ((kw == 0) ? (1.0f / 4096.0f) : (2.0f / 4096.0f));
}

// ---------------------------------------------------------------------------
// Fused DCT -> filter -> inverse chain.  One block per (batch, 2-channel chunk).
// LDS: sTP3 34KB + sTP4 33KB + sA 34KB + sB 68KB ~= 169KB (of 320KB/WGP).
#define CC 2
__global__ __launch_bounds__(512) void dct_chain_kernel(
    const float* __restrict__ x, const float* __restrict__ ws,
    float* __restrict__ xout) {
  __shared__ __align__(16) float sTP3[128 * S64];   // [m][k] padded
  __shared__ __align__(16) float sTP4[64 * S128];   // [j][k2] padded
  __shared__ __align__(16) float sA[128 * S64];     // multi-layout ping buffer
  __shared__ __align__(16) float sB[256 * S64];     // pong buffer

  const int b  = blockIdx.x >> 7;           // 8 batches
  const int cb = (blockIdx.x & 127) * CC;   // channel base
  const float* TP3 = ws;
  const float* TP4 = ws + 8192;
  const float* Msc = ws + 16384;
  const int tid = threadIdx.x;

  for (int i = tid; i < 8192; i += 512) {
    sTP3[(i >> 6) * S64 + (i & 63)] = TP3[i];
    sTP4[(i >> 7) * S128 + (i & 127)] = TP4[i];
  }

  // Stage 0: load x tile -> sA row-major [i][j*2+c] (flat, contiguous 32KB).
  const float* xb = x + (size_t)b * 4096 * 256 + cb;
#if defined(__HIP_DEVICE_COMPILE__) && \
    __has_builtin(__builtin_amdgcn_tensor_load_to_lds)
  if (tid < 32) {
    // Tensor Data Mover: 2D tile, dim0 = 2 f32 (channels), dim1 = 4096 pixels,
    // row stride 256 f32 -> fills LDS contiguously in [pixel][c] order.
    unsigned lds_off = (unsigned)(unsigned long long)(const void*)sA;
    unsigned long long ga = (unsigned long long)xb;
    u32x4 g0;
    g0[0] = 1u;                                   // count=1 (valid descriptor)
    g0[1] = lds_off;                              // lds_addr (bytes)
    g0[2] = (unsigned)(ga & 0xffffffffu);         // global_addr[31:0]
    g0[3] = (unsigned)((ga >> 32) & 0x1ffffffu)   // global_addr[56:32]
            | 0x80000000u;                        // type=2 ("image")
    i32x8 g1;
    g1[0] = 0x20000;       // data_size=2 (4 bytes) at bits [17:16]
    g1[1] = 0x20000;       // tensor_dim0 = 2     at bits [63:48]
    g1[2] = 0x10000000;    // tensor_dim1 = 4096  at bits [111:80]
    g1[3] = 0x20000;       // tile_dim0   = 2     at bits [127:112]
    g1[4] = 0x1000;        // tile_dim1   = 4096  at bits [143:128]
    g1[5] = 256;           // tensor_dim0_stride = 256 at bits [207:160]
    g1[6] = 0;
    g1[7] = 0;
    i32x4 z4 = {0, 0, 0, 0};
    i32x8 z8 = {0, 0, 0, 0, 0, 0, 0, 0};
    __builtin_amdgcn_tensor_load_to_lds(g0, g1, z4, z4, z8, 0);
    __builtin_amdgcn_s_wait_tensorcnt(0);
  }
#else
  for (int p = tid; p < 4096; p += 512) {
    float2 v = *(const float2*)(xb + (size_t)p * 256);
    sA[p * 2 + 0] = v.x;
    sA[p * 2 + 1] = v.y;
  }
#endif
  __syncthreads();

  // P1: out[kh][j,c] = sum_i Tc[kh][i] * X[i][j,c]
  //     A[m=kh][k=i] = Tc[kh][i] = sTP3[k][m]   (transposed read, lane-consec.)
  //     B row-major: sA[k*128 + n]              (lane-consecutive, no conflict)
  //     store -> sB col-major [(kh*2+c)][j]
  wmma_pass<2, 8>(4, 8, 16,
      [&](int m, int k0) {
        v2f a; a.x = sTP3[k0 * S64 + m]; a.y = sTP3[(k0 + 1) * S64 + m];
        return a;
      },
      [&](int k0, int n) {
        v2f bv; bv.x = sA[k0 * 128 + n]; bv.y = sA[(k0 + 1) * 128 + n];
        return bv;
      },
      [&](int m, int n, float v) { sB[(m * 2 + (n & 1)) * S64 + (n >> 1)] = v; });
  __syncthreads();

  // P2: E[kh][kw][c] = sum_j Tc[kw][j] * P1out[kh][j][c]
  //     rows m=kw, cols n=(kh*2+c); B k-contiguous -> ds_load_b64
  //     store -> sA col-major [(kw*2+c)][kh]
  wmma_pass<2, 8>(4, 8, 16,
      [&](int m, int k0) {
        v2f a; a.x = sTP3[k0 * S64 + m]; a.y = sTP3[(k0 + 1) * S64 + m];
        return a;
      },
      [&](int k0, int n) { return *(const v2f*)&sB[n * S64 + k0]; },
      [&](int m, int n, float v) { sA[(m * 2 + (n & 1)) * S64 + (n >> 1)] = v; });
  __syncthreads();

  // Elementwise band filter (g_kw/4096 folded into Msc)
  for (int idx = tid; idx < 8192; idx += 512) {
    int n3 = idx >> 6, kh = idx & 63;
    int kw = n3 >> 1, c = n3 & 1;
    sA[n3 * S64 + kh] *= Msc[(((kh << 6) + kw) << 8) + cb + c];
  }
  __syncthreads();

  // P3: F[m][kw][c] = sum_kh TP3[m][kh] * E[kh][kw][c]   (m<64: Fc, m>=64: Fs)
  //     A row-major contiguous (padded, conflict-free b64); B k-contiguous b64
  //     store -> sB [(i*2+c) + 128*(m>=64)][kw]
  wmma_pass<4, 8>(8, 8, 16,
      [&](int m, int k0) { return *(const v2f*)&sTP3[m * S64 + k0]; },
      [&](int k0, int n) { return *(const v2f*)&sA[n * S64 + k0]; },
      [&](int m, int n, float v) {
        sB[(((m & 63) * 2 + (n & 1)) + ((m >> 6) << 7)) * S64 + (n >> 1)] = v;
      });
  __syncthreads();

  // P4: xo[i][j][c] = sum_{k2<128} TP4[j][k2] * H[k2][i][c]
  //     H[k2][i][c] = sB[(i*2+c) + 128*(k2>=64)][k2&63]; pair never crosses 64
  //     store -> sA final [i][j][c] with padded pixel-row stride 130
  wmma_pass<2, 8>(4, 8, 32,
      [&](int m, int k0) { return *(const v2f*)&sTP4[m * S128 + k0]; },
      [&](int k0, int n) {
        return *(const v2f*)&sB[(n + ((k0 & 64) << 1)) * S64 + (k0 & 63)];
      },
      [&](int m, int n, float v) { sA[(n >> 1) * 130 + m * 2 + (n & 1)] = v; });
  __syncthreads();

  float* ob = xout + (size_t)b * 4096 * 256 + cb;
  for (int p = tid; p < 4096; p += 512) {
    int i = p >> 6, j = p & 63;
    float2 v;
    v.x = sA[i * 130 + j * 2 + 0];
    v.y = sA[i * 130 + j * 2 + 1];
    *(float2*)(ob + (size_t)p * 256) = v;
  }
}

// ---------------------------------------------------------------------------
// Projection GEMM (y = xo · pw^T) + fused LayerNorm, in place on d_out.
// Each block owns 64 rows exclusively -> safe read-then-overwrite.
// MGROUP=4: each proj_w fragment is loaded once per block (256KB L2 traffic).
__global__ __launch_bounds__(512) void proj_ln_kernel(
    float* __restrict__ y, const float* __restrict__ pw,
    const float* __restrict__ g, const float* __restrict__ bb) {
  __shared__ __align__(16) float sX[64 * S256];  // padded input rows (~65KB)
  __shared__ __align__(16) float sY[64 * 256];   // projected rows (64KB)
  __shared__ float sMu[64], sRs[64];
  const int r0 = blockIdx.x * 64;
  const int tid = threadIdx.x;

  // Warm proj_w into L2 (global_prefetch_b8), 256KB, fully L2-resident.
  for (int q = tid; q < 2048; q += 512) __builtin_prefetch(pw + q * 32, 0, 1);

  const float4* src = (const float4*)(y + (size_t)r0 * 256);
  for (int p = tid; p < 64 * 256 / 4; p += 512)
    *(float4*)(sX + (p >> 6) * S256 + ((p & 63) << 2)) = src[p];
  __syncthreads();

  // M=64 rows, N=256 out channels, K=256. A from LDS (b64, conflict-free
  // thanks to S256 % 64 == 4), B (proj_w) straight from L2 as b64.
  wmma_pass<4, 8>(4, 16, 64,
      [&](int m, int k0) { return *(const v2f*)&sX[m * S256 + k0]; },
      [&](int k0, int n) { return *(const v2f*)&pw[n * 256 + k0]; },
      [&](int m, int n, float v) { sY[m * 256 + n] = v; });
  __syncthreads();

  // LayerNorm stats: 8 lanes per row (groups live inside one wave32)
  {
    int row = tid >> 3, sub = tid & 7;
    float s = 0.f, s2 = 0.f;
    for (int c = sub; c < 256; c += 8) {
      float v = sY[row * 256 + c];
      s += v;
      s2 += v * v;
    }
    s  += __shfl_xor(s, 1);  s  += __shfl_xor(s, 2);  s  += __shfl_xor(s, 4);
    s2 += __shfl_xor(s2, 1); s2 += __shfl_xor(s2, 2); s2 += __shfl_xor(s2, 4);
    if (sub == 0) {
      float mu = s * (1.0f / 256.0f);
      float var = s2 * (1.0f / 256.0f) - mu * mu;
      sMu[row] = mu;
      sRs[row] = rsqrtf(var + 1e-5f);
    }
  }
  __syncthreads();

  for (int p = tid; p < 64 * 256; p += 512) {
    int row = p >> 8, d = p & 255;
    y[(size_t)r0 * 256 + p] = (sY[p] - sMu[row]) * sRs[row] * g[d] + bb[d];
  }
}

// ---------------------------------------------------------------------------
extern "C" void kernel_launch(void* const* d_in, const int* in_sizes, int n_in,
                              void* d_out, int out_size, void* d_ws, size_t ws_size,
                              hipStream_t stream) {
  const float* x  = (const float*)d_in[0];
  const float* Wl = (const float*)d_in[1];
  const float* Wm = (const float*)d_in[2];
  const float* Wh = (const float*)d_in[3];
  const float* pw = (const float*)d_in[4];
  const float* g  = (const float*)d_in[5];
  const float* bb = (const float*)d_in[6];
  float* out = (float*)d_out;
  float* ws  = (float*)d_ws;   // [0,16384): TP3|TP4, [16384, +64*64*256): Msc

  prep_T_kernel<<<64, 256, 0, stream>>>(ws);
  prep_filter_kernel<<<4096, 256, 0, stream>>>(Wl, Wm, Wh, ws + 16384);
  dct_chain_kernel<<<8 * 128, 512, 0, stream>>>(x, ws, out);
  proj_ln_kernel<<<512, 512, 0, stream>>>(out, pw, g, bb);
}